// Embedding_75153337745818
// MI455X (gfx1250) — compile-verified
//
#include <hip/hip_runtime.h>
#include <hip/hip_bf16.h>

// Embedding row gather: out[i, :] = table[ids[i], :]
// table: [NUM_EMBEDDINGS, 64] f32, ids: [NUM_IDS] i32, out: [NUM_IDS, 64] f32.
//
// Memory-bound: ~1.03 GB total traffic -> ~44 us floor at 23.3 TB/s HBM.
//   - 16 lanes per row, one 128-bit load + one 128-bit store per lane
//     (row = 256 B = 16 x b128). Wave32 covers 2 rows -> two contiguous
//     256 B load bursts + one coalesced 512 B store per wave.
//   - Table loads: default (RT) temporal policy -> ~18% duplicate ids can
//     hit in the 192 MB L2.
//   - Output stores: non-temporal (gfx1250 TH_NT) -> the write-once 512 MB
//     output stream does not evict reusable table rows from L2.
//   - dim4 hardcoded to 16 in the fast path so address math is pure
//     shift+add (v_lshl_add_u64) instead of 64-bit multiplies.

typedef __attribute__((ext_vector_type(4))) float v4f;

__global__ __launch_bounds__(256) void Embedding_75153337745818_kernel_d16(
    const v4f* __restrict__ table4,   // [NUM_EMBEDDINGS][16] float4
    const int* __restrict__ ids,      // [n_ids] int32 row indices
    v4f* __restrict__ out4,           // [n_ids][16] float4
    unsigned n_ids)
{
    // One thread moves one 16-byte chunk of one row.
    const unsigned t     = blockIdx.x * 256u + threadIdx.x;   // < 32M, fits u32
    const unsigned row   = t >> 4;                            // 16 lanes per row
    const unsigned chunk = t & 15u;

    if (row >= n_ids) return;

    // Lanes 0-15 of a wave share one id, lanes 16-31 the next; duplicate id
    // loads coalesce to one transaction per half-wave.
    const long long id = (long long)ids[row];

    // Table byte offset = id * 256 -> single v_lshl_add_u64 (shift 8).
    const v4f v = table4[id * 16 + chunk];

    // Output byte offset = row * 256 + chunk * 16 (< 512 MB, fits u32 scale).
    __builtin_nontemporal_store(v, &out4[(size_t)row * 16 + chunk]);
}

// Generic fallback for dim % 4 == 0, dim != 64 (not expected for this ref).
__global__ __launch_bounds__(256) void Embedding_75153337745818_kernel_gen(
    const v4f* __restrict__ table4,
    const int* __restrict__ ids,
    v4f* __restrict__ out4,
    unsigned n_ids,
    unsigned dim4)
{
    const unsigned long long t = (unsigned long long)blockIdx.x * 256u + threadIdx.x;
    const unsigned long long row = t / dim4;
    const unsigned chunk = (unsigned)(t % dim4);
    if (row >= n_ids) return;
    const long long id = (long long)ids[row];
    const v4f v = table4[id * (long long)dim4 + chunk];
    __builtin_nontemporal_store(v, &out4[row * dim4 + chunk]);
}

extern "C" void kernel_launch(void* const* d_in, const int* in_sizes, int n_in,
                              void* d_out, int out_size, void* d_ws, size_t ws_size,
                              hipStream_t stream) {
    const float* table = (const float*)d_in[0];
    const int*   ids   = (const int*)d_in[1];
    float*       out   = (float*)d_out;

    const unsigned n_ids = (unsigned)in_sizes[1];               // 2,000,000
    const unsigned dim   = n_ids ? (unsigned)(out_size / (int)n_ids) : 0;  // 64
    const unsigned dim4  = dim >> 2;                            // 16

    const int block = 256;                                      // 8 wave32 / block

    if (dim4 == 16u) {
        // One thread per 16-byte chunk: n_ids * 16 threads (32M).
        const unsigned long long total = (unsigned long long)n_ids * 16u;
        const unsigned grid = (unsigned)((total + block - 1) / block);
        Embedding_75153337745818_kernel_d16<<<dim3(grid), dim3(block), 0, stream>>>(
            (const v4f*)table, ids, (v4f*)out, n_ids);
    } else if (dim4 > 0) {
        const unsigned long long total = (unsigned long long)n_ids * dim4;
        const unsigned grid = (unsigned)((total + block - 1) / block);
        Embedding_75153337745818_kernel_gen<<<dim3(grid), dim3(block), 0, stream>>>(
            (const v4f*)table, ids, (v4f*)out, n_ids, dim4);
    }
}